// semidual_32504312496602
// MI455X (gfx1250) — compile-verified
//
#include <hip/hip_runtime.h>
#include <hip/hip_bf16.h>

typedef __attribute__((ext_vector_type(2))) float v2f;
typedef __attribute__((ext_vector_type(8))) float v8f;

#define N_ROWS 8192
#define M_COLS 32768
#define DDIM   64
#define NCHUNK 16                         // column-split factor
#define TILES_PER_CHUNK (M_COLS / 16 / NCHUNK)   // 128 column tiles per block
#define FLT_BIG 3.402823466e+38f

// ---------------------------------------------------------------------------
// Kernel 1: t[j] = ||y_j||^2 - psi[j]      (M elements)
// ---------------------------------------------------------------------------
__global__ void prep_t_kernel(const float* __restrict__ Y,
                              const float* __restrict__ psi,
                              float* __restrict__ t) {
    const int j = blockIdx.x * 256 + threadIdx.x;
    const float4* yp = (const float4*)(Y + (size_t)j * DDIM);
    float s = 0.0f;
#pragma unroll
    for (int k = 0; k < DDIM / 4; ++k) {
        float4 v = yp[k];
        s += v.x * v.x + v.y * v.y + v.z * v.z + v.w * v.w;
    }
    t[j] = s - psi[j];
}

// ---------------------------------------------------------------------------
// Kernel 2: partial[chunk][i] = min_{j in chunk} ( t[j] - 2 * x_i . y_j )
//
// WMMA f32 16x16x4. A fragment (16x4): lane L holds row L&15, K pair
// {4s+koff, 4s+koff+1}, koff = 2*(L>=16). B fragment (4x16): same lane
// mapping with "row" = column j. C/D: VGPR v, lane L -> row v+8*(L>>4),
// col L&15.
//
// Block = 128 threads (4 waves); each wave owns two 16-row tiles (A held in
// registers, pre-scaled by -2). blockIdx.x selects the 128-row group,
// blockIdx.y the 2048-column chunk. B fragments are double-buffered: tile
// ct+1 is prefetched (16 b64 loads issued as one clause) while tile ct's 32
// WMMAs execute.
// ---------------------------------------------------------------------------
__device__ __forceinline__ void load_b_frag(v2f b[16], const float* __restrict__ Y,
                                            int j, int koff) {
    const float* yp = Y + (size_t)j * DDIM + koff;
#pragma unroll
    for (int s = 0; s < 16; ++s) b[s] = *(const v2f*)(yp + 4 * s);
}

__global__ void __launch_bounds__(128)
minred_kernel(const float* __restrict__ X,
              const float* __restrict__ Y,
              const float* __restrict__ t,
              float* __restrict__ partial) {
    const int wave = threadIdx.x >> 5;
    const int lane = threadIdx.x & 31;
    const int half = lane >> 4;        // 0: K%4 in {0,1}; 1: K%4 in {2,3}
    const int lrow = lane & 15;
    const int koff = half * 2;
    const int rowbase = blockIdx.x * 128 + wave * 32;     // 32 rows per wave
    const int colbase = blockIdx.y * (TILES_PER_CHUNK * 16);

    // Preload A fragments for both 16-row tiles, pre-scaled by -2.
    v2f a0[16], a1[16];
#pragma unroll
    for (int s = 0; s < 16; ++s) {
        const float* p0 = X + (size_t)(rowbase + lrow) * DDIM + 4 * s + koff;
        const float* p1 = X + (size_t)(rowbase + 16 + lrow) * DDIM + 4 * s + koff;
        a0[s] = (*(const v2f*)p0) * -2.0f;
        a1[s] = (*(const v2f*)p1) * -2.0f;
    }

    float r0[8], r1[8];
#pragma unroll
    for (int v = 0; v < 8; ++v) { r0[v] = FLT_BIG; r1[v] = FLT_BIG; }

    // Software-pipelined loop over column tiles (double-buffered B).
    v2f bA[16], bB[16];
    const int j0 = colbase + lrow;
    load_b_frag(bA, Y, j0, koff);

    for (int ct = 0; ct < TILES_PER_CHUNK; ct += 2) {
        const int jcur  = colbase + ct * 16 + lrow;
        // prefetch tile ct+1 (always in-range: TILES_PER_CHUNK is even)
        load_b_frag(bB, Y, jcur + 16, koff);

        {
            const float tv = t[jcur];
            v8f c0 = {tv, tv, tv, tv, tv, tv, tv, tv};
            v8f c1 = c0;
#pragma unroll
            for (int s = 0; s < 16; ++s) {
                c0 = __builtin_amdgcn_wmma_f32_16x16x4_f32(
                    false, a0[s], false, bA[s], (short)0, c0, false, false);
                c1 = __builtin_amdgcn_wmma_f32_16x16x4_f32(
                    false, a1[s], false, bA[s], (short)0, c1, false, false);
            }
#pragma unroll
            for (int v = 0; v < 8; ++v) {
                r0[v] = fminf(r0[v], c0[v]);
                r1[v] = fminf(r1[v], c1[v]);
            }
        }

        // prefetch tile ct+2 (clamped so the last iteration stays in-range)
        const int jpre = (ct + 2 < TILES_PER_CHUNK)
                             ? (colbase + (ct + 2) * 16 + lrow)
                             : (colbase + lrow);
        load_b_frag(bA, Y, jpre, koff);

        {
            const float tv = t[jcur + 16];
            v8f c0 = {tv, tv, tv, tv, tv, tv, tv, tv};
            v8f c1 = c0;
#pragma unroll
            for (int s = 0; s < 16; ++s) {
                c0 = __builtin_amdgcn_wmma_f32_16x16x4_f32(
                    false, a0[s], false, bB[s], (short)0, c0, false, false);
                c1 = __builtin_amdgcn_wmma_f32_16x16x4_f32(
                    false, a1[s], false, bB[s], (short)0, c1, false, false);
            }
#pragma unroll
            for (int v = 0; v < 8; ++v) {
                r0[v] = fminf(r0[v], c0[v]);
                r1[v] = fminf(r1[v], c1[v]);
            }
        }
    }

    // Cross-lane min within each 16-lane half (masks < 16 stay in-half).
    float* out = partial + (size_t)blockIdx.y * N_ROWS;
#pragma unroll
    for (int v = 0; v < 8; ++v) {
        float a = r0[v];
        float b = r1[v];
#pragma unroll
        for (int m = 1; m <= 8; m <<= 1) {
            a = fminf(a, __shfl_xor(a, m, 32));
            b = fminf(b, __shfl_xor(b, m, 32));
        }
        if (lrow == 0) {
            const int rr = rowbase + half * 8 + v;
            out[rr]      = a;   // tile 0 (rows rowbase .. +15)
            out[rr + 16] = b;   // tile 1 (rows rowbase+16 .. +31)
        }
    }
}

// ---------------------------------------------------------------------------
// Kernel 3: out = mean_i( min_k partial[k][i] + ||x_i||^2 ) + sum(w*psi)/sum(w)
// Single block, deterministic tree reduction.
// ---------------------------------------------------------------------------
__global__ void __launch_bounds__(512)
finalize_kernel(const float* __restrict__ X,
                const float* __restrict__ partial,
                const float* __restrict__ psi,
                const float* __restrict__ w,
                float* __restrict__ out) {
    __shared__ float s1[512];
    __shared__ float s2[512];
    __shared__ float s3[512];
    const int tid = threadIdx.x;

    float acc = 0.0f;
    for (int i = tid; i < N_ROWS; i += 512) {
        float rm = FLT_BIG;
#pragma unroll
        for (int k = 0; k < NCHUNK; ++k) rm = fminf(rm, partial[(size_t)k * N_ROWS + i]);
        const float4* xp = (const float4*)(X + (size_t)i * DDIM);
        float x2 = 0.0f;
#pragma unroll
        for (int k = 0; k < DDIM / 4; ++k) {
            float4 v = xp[k];
            x2 += v.x * v.x + v.y * v.y + v.z * v.z + v.w * v.w;
        }
        acc += rm + x2;
    }

    float wp = 0.0f, ws = 0.0f;
    for (int j = tid; j < M_COLS; j += 512) {
        const float ww = w[j];
        wp += ww * psi[j];
        ws += ww;
    }

    s1[tid] = acc; s2[tid] = wp; s3[tid] = ws;
    __syncthreads();
    for (int off = 256; off > 0; off >>= 1) {
        if (tid < off) {
            s1[tid] += s1[tid + off];
            s2[tid] += s2[tid + off];
            s3[tid] += s3[tid + off];
        }
        __syncthreads();
    }
    if (tid == 0) {
        out[0] = s1[0] / (float)N_ROWS + s2[0] / s3[0];
    }
}

// ---------------------------------------------------------------------------
extern "C" void kernel_launch(void* const* d_in, const int* in_sizes, int n_in,
                              void* d_out, int out_size, void* d_ws, size_t ws_size,
                              hipStream_t stream) {
    const float* x   = (const float*)d_in[0];   // [N, 64]
    const float* y   = (const float*)d_in[1];   // [M, 64]
    const float* psi = (const float*)d_in[2];   // [M]
    const float* w   = (const float*)d_in[3];   // [M]
    float* out = (float*)d_out;

    float* t_ws   = (float*)d_ws;               // [M]          t_j
    float* part   = t_ws + M_COLS;              // [NCHUNK][N]  partial minima

    prep_t_kernel<<<M_COLS / 256, 256, 0, stream>>>(y, psi, t_ws);
    dim3 grid(N_ROWS / 128, NCHUNK);
    minred_kernel<<<grid, 128, 0, stream>>>(x, y, t_ws, part);
    finalize_kernel<<<1, 512, 0, stream>>>(x, part, psi, w, out);
}